// CollisionRegularizer_22359599743152
// MI455X (gfx1250) — compile-verified
//
#include <hip/hip_runtime.h>
#include <math.h>

#define NPTS   2048
#define NBATCH 4
#define NTILES (NPTS / 16)            // 128 column tiles per row strip
#define NBLK   (NBATCH * NTILES)      // 512 blocks = (batch, row-strip)

typedef __attribute__((ext_vector_type(2))) float v2f;
typedef __attribute__((ext_vector_type(8))) float v8f;

__device__ __forceinline__ void quat_to_R(float qw, float qx, float qy, float qz,
                                          float R[9]) {
    R[0] = 1.f - 2.f * (qy * qy + qz * qz);
    R[1] = 2.f * (qx * qy - qz * qw);
    R[2] = 2.f * (qx * qz + qy * qw);
    R[3] = 2.f * (qx * qy + qz * qw);
    R[4] = 1.f - 2.f * (qx * qx + qz * qz);
    R[5] = 2.f * (qy * qz - qx * qw);
    R[6] = 2.f * (qx * qz - qy * qw);
    R[7] = 2.f * (qy * qz + qx * qw);
    R[8] = 1.f - 2.f * (qx * qx + qy * qy);
}

// One block = (batch b, 16-row strip I). 8 waves sweep the 128 column tiles.
__global__ __launch_bounds__(256) void collision_tiles(
    const float* __restrict__ xyz, const float* __restrict__ scales,
    const float* __restrict__ rot, const float* __restrict__ vel,
    float* __restrict__ partials) {
    __shared__ float sX[16][3], sV[16][3], sR[16][9], sP[16][3], sS2[16][3];
    __shared__ float sXX[16], sCC[16];
    __shared__ float red[256];

    const int b    = blockIdx.x >> 7;
    const int I    = (blockIdx.x & 127) << 4;
    const int tid  = threadIdx.x;
    const int lane = tid & 31;
    const int wid  = tid >> 5;
    const int lo   = lane & 15;
    const bool hi  = lane >= 16;

    // ---- row-strip precompute: R, p = x^T R, s^2, |x|^2, v.x ----
    if (tid < 16) {
        const int base = (b * NPTS + I + tid);
        const float x0 = xyz[base * 3 + 0], x1 = xyz[base * 3 + 1], x2 = xyz[base * 3 + 2];
        const float qw = rot[base * 4 + 0], qx = rot[base * 4 + 1];
        const float qy = rot[base * 4 + 2], qz = rot[base * 4 + 3];
        const float s0 = scales[base * 3 + 0], s1 = scales[base * 3 + 1], s2 = scales[base * 3 + 2];
        const float w0 = vel[base * 3 + 0], w1 = vel[base * 3 + 1], w2 = vel[base * 3 + 2];
        float R[9];
        quat_to_R(qw, qx, qy, qz, R);
#pragma unroll
        for (int k = 0; k < 9; k++) sR[tid][k] = R[k];
        sX[tid][0] = x0; sX[tid][1] = x1; sX[tid][2] = x2;
        sV[tid][0] = w0; sV[tid][1] = w1; sV[tid][2] = w2;
        sS2[tid][0] = s0 * s0; sS2[tid][1] = s1 * s1; sS2[tid][2] = s2 * s2;
        sP[tid][0] = x0 * R[0] + x1 * R[3] + x2 * R[6];
        sP[tid][1] = x0 * R[1] + x1 * R[4] + x2 * R[7];
        sP[tid][2] = x0 * R[2] + x1 * R[5] + x2 * R[8];
        sXX[tid] = x0 * x0 + x1 * x1 + x2 * x2;
        sCC[tid] = w0 * x0 + w1 * x1 + w2 * x2;
    }
    __syncthreads();

    // Wave-invariant WMMA A operands.
    // A 16x4 layout: lanes 0-15 hold K=0(v0)/K=1(v1); lanes 16-31 hold K=2(v0)/K=3(v1, =0 pad)
    v2f ax, av, aR[3];
    ax.x = hi ? sX[lo][2] : sX[lo][0];
    ax.y = hi ? 0.f       : sX[lo][1];
    av.x = hi ? sV[lo][2] : sV[lo][0];
    av.y = hi ? 0.f       : sV[lo][1];
#pragma unroll
    for (int j = 0; j < 3; j++) {             // A rows = R_n[:,j]
        aR[j].x = hi ? sR[lo][6 + j] : sR[lo][j];
        aR[j].y = hi ? 0.f           : sR[lo][3 + j];
    }

    float acc = 0.f;

#pragma unroll 1   // no software-pipelining of two tiles (register pressure)
    for (int t = wid; t < NTILES; t += 8) {
        const int J  = t << 4;
        const int mb = b * NPTS + J + lo;

        // column-point data (recomputed in-register; L2-resident loads)
        const float xm0 = xyz[mb * 3 + 0], xm1 = xyz[mb * 3 + 1], xm2 = xyz[mb * 3 + 2];
        const float vm0 = vel[mb * 3 + 0], vm1 = vel[mb * 3 + 1], vm2 = vel[mb * 3 + 2];
        const float qw = rot[mb * 4 + 0], qx = rot[mb * 4 + 1];
        const float qy = rot[mb * 4 + 2], qz = rot[mb * 4 + 3];
        float Rm[9];
        quat_to_R(qw, qx, qy, qz, Rm);
        const float t0 = scales[mb * 3 + 0], t1s = scales[mb * 3 + 1], t2s = scales[mb * 3 + 2];
        float s2m[3] = {t0 * t0, t1s * t1s, t2s * t2s};
        float pm[3];
        pm[0] = xm0 * Rm[0] + xm1 * Rm[3] + xm2 * Rm[6];
        pm[1] = xm0 * Rm[1] + xm1 * Rm[4] + xm2 * Rm[7];
        pm[2] = xm0 * Rm[2] + xm1 * Rm[5] + xm2 * Rm[8];
        const float xxm = xm0 * xm0 + xm1 * xm1 + xm2 * xm2;
        const float ccm = vm0 * xm0 + vm1 * xm1 + vm2 * xm2;

        // WMMA B operands (column side): x, v, and R_m[:,j]
        v2f bx, bv, bR[3];
        bx.x = hi ? xm2 : xm0;
        bx.y = hi ? 0.f : xm1;
        bv.x = hi ? vm2 : vm0;
        bv.y = hi ? 0.f : vm1;
#pragma unroll
        for (int j = 0; j < 3; j++) {         // B cols = R_m[:,j]
            bR[j].x = hi ? Rm[6 + j] : Rm[j];
            bR[j].y = hi ? 0.f       : Rm[3 + j];
        }

        v8f cz = {};
        // Gram matrices on the matrix pipe:
        //   gxx  = x_n . x_m
        //   gsum = v_n . x_m + x_n . v_m   (chained through C)
        //   t1_j = x_m . R_n[:,j]          (row rotation applied to direction)
        //   t2_j = x_n . R_m[:,j]          (col rotation applied to direction)
        v8f gxx  = __builtin_amdgcn_wmma_f32_16x16x4_f32(false, ax, false, bx, (short)0, cz,  false, false);
        v8f gxv  = __builtin_amdgcn_wmma_f32_16x16x4_f32(false, ax, false, bv, (short)0, cz,  false, false);
        v8f gsum = __builtin_amdgcn_wmma_f32_16x16x4_f32(false, av, false, bx, (short)0, gxv, false, false);
        v8f tt1[3], tt2[3];
#pragma unroll
        for (int j = 0; j < 3; j++) {
            tt1[j] = __builtin_amdgcn_wmma_f32_16x16x4_f32(false, aR[j], false, bx,    (short)0, cz, false, false);
            tt2[j] = __builtin_amdgcn_wmma_f32_16x16x4_f32(false, ax,    false, bR[j], (short)0, cz, false, false);
        }

#pragma unroll
        for (int v = 0; v < 8; v++) {
            const int r = v + (hi ? 8 : 0);  // C/D layout: M = v + 8*(lane>=16)
            const float dist2 = sXX[r] + xxm - 2.f * gxx[v] + 1e-8f;
            const float inv   = rsqrtf(dist2);
            const float dist  = dist2 * inv;

            float rr1 = 0.f, rr2 = 0.f;
#pragma unroll
            for (int j = 0; j < 3; j++) {
                const float u1 = sP[r][j] - tt1[j][v];   // r_dir[n,m], scaled by s_m
                rr1 += s2m[j] * u1 * u1;
                const float u2 = tt2[j][v] - pm[j];      // r_dir[m,n], scaled by s_n
                rr2 += sS2[r][j] * u2 * u2;
            }
            const float rtot = (sqrtf(rr1) + sqrtf(rr2)) * inv;
            const float ov   = fmaxf(rtot - dist, 0.f);
            const float sp   = ov * ov / (1.f + 0.1f * ov);
            const float negva = (gsum[v] - sCC[r] - ccm) * inv;  // -v_approach
            acc += sp + 0.1f * ov * fmaxf(negva, 0.f);
        }
    }

    // deterministic block tree-reduce -> one partial per block
    red[tid] = acc;
    __syncthreads();
    for (int s = 128; s > 0; s >>= 1) {
        if (tid < s) red[tid] += red[tid + s];
        __syncthreads();
    }
    if (tid == 0) partials[blockIdx.x] = red[0];
}

__global__ __launch_bounds__(256) void final_reduce(const float* __restrict__ partials,
                                                    float* __restrict__ out) {
    __shared__ float red[256];
    const int tid = threadIdx.x;
    red[tid] = partials[tid] + partials[tid + 256];  // 512 partials
    __syncthreads();
    for (int s = 128; s > 0; s >>= 1) {
        if (tid < s) red[tid] += red[tid + s];
        __syncthreads();
    }
    if (tid == 0)
        out[0] = red[0] * (1.0f / ((float)NBATCH * (float)NPTS * (float)NPTS));
}

extern "C" void kernel_launch(void* const* d_in, const int* in_sizes, int n_in,
                              void* d_out, int out_size, void* d_ws, size_t ws_size,
                              hipStream_t stream) {
    const float* xyz    = (const float*)d_in[0];
    const float* scales = (const float*)d_in[1];
    const float* rot    = (const float*)d_in[2];
    const float* vel    = (const float*)d_in[3];
    float* partials     = (float*)d_ws;   // 512 floats
    float* out          = (float*)d_out;  // single f32 scalar

    collision_tiles<<<NBLK, 256, 0, stream>>>(xyz, scales, rot, vel, partials);
    final_reduce<<<1, 256, 0, stream>>>(partials, out);
}